// MultiHeadCrossAttention_72971494359383
// MI455X (gfx1250) — compile-verified
//
#include <hip/hip_runtime.h>
#include <math.h>

#define B_  2
#define S1_ 1024
#define S2_ 1024
#define D_  64
#define H_  16
#define DH_ 1024

typedef __attribute__((ext_vector_type(2))) float v2f;
typedef __attribute__((ext_vector_type(8))) float v8f;

__device__ __forceinline__ v8f wmma_f32(v2f a, v2f b, v8f c) {
  // 8 args: (neg_a, A, neg_b, B, c_mod, C, reuse_a, reuse_b)
  return __builtin_amdgcn_wmma_f32_16x16x4_f32(false, a, false, b, (short)0, c,
                                               false, false);
}

// Y(MxN) = X(MxK) @ W(KxN) + bias(N). One wave per 32x32 block (2x2 tiles).
// N, K compile-time so all loads use immediate offsets.
template <int N, int K>
__global__ void gemm_bias_kernel(const float* __restrict__ X,
                                 const float* __restrict__ W,
                                 const float* __restrict__ bias,
                                 float* __restrict__ Y) {
  const int lane = threadIdx.x;          // 0..31 (wave32)
  const int r    = lane & 15;
  const int kh   = lane >> 4;            // K-half select (0/1)
  const int m0   = blockIdx.x * 32;
  const int n0   = (blockIdx.y * blockDim.y + threadIdx.y) * 32;
  const float* xr0 = X + (size_t)(m0 + r) * K + 2 * kh;   // A rows (lane r = M)
  const float* xr1 = xr0 + (size_t)16 * K;
  const float* w0  = W + (size_t)(2 * kh) * N + n0 + r;   // B cols (lane r = N)
  v8f c00 = {}, c01 = {}, c10 = {}, c11 = {};
#pragma unroll 8
  for (int kb = 0; kb < K; kb += 4) {
    const v2f a0 = *(const v2f*)(xr0 + kb);
    const v2f a1 = *(const v2f*)(xr1 + kb);
    v2f b0, b1;
    b0.x = w0[kb * N];
    b0.y = w0[kb * N + N];
    b1.x = w0[kb * N + 16];
    b1.y = w0[kb * N + N + 16];
    c00 = wmma_f32(a0, b0, c00);
    c01 = wmma_f32(a0, b1, c01);
    c10 = wmma_f32(a1, b0, c10);
    c11 = wmma_f32(a1, b1, c11);
  }
  const float bv0 = bias[n0 + r];
  const float bv1 = bias[n0 + 16 + r];
  float* y0 = Y + (size_t)(m0 + kh * 8) * N + n0 + r;     // rows m0+v+8*kh
  float* y1 = y0 + (size_t)16 * N;                        // rows +16
#pragma unroll
  for (int v = 0; v < 8; ++v) {
    y0[(size_t)v * N]      = c00[v] + bv0;
    y0[(size_t)v * N + 16] = c01[v] + bv1;
    y1[(size_t)v * N]      = c10[v] + bv0;
    y1[(size_t)v * N + 16] = c11[v] + bv1;
  }
}

// scores[bh,i,j] = sum_d Q[b,i,h,d] * K[b,j,h,d]; |s| always, signed optionally.
template <bool WRITE_SIGNED>
__global__ void scores_kernel(const float* __restrict__ Qf,
                              const float* __restrict__ Kf,
                              float* __restrict__ sgn,
                              float* __restrict__ abs_o) {
  const int lane = threadIdx.x;
  const int r    = lane & 15;
  const int kh   = lane >> 4;
  const int bh   = blockIdx.z;
  const int b    = bh >> 4;              // / H_
  const int h    = bh & 15;
  const int i0   = blockIdx.x * 32;
  const int j0   = (blockIdx.y * blockDim.y + threadIdx.y) * 32;
  const float* q0 = Qf + (size_t)(b * S1_ + i0 + r) * DH_ + h * D_ + 2 * kh;
  const float* q1 = q0 + (size_t)16 * DH_;
  const float* k0 = Kf + (size_t)(b * S2_ + j0 + r) * DH_ + h * D_ + 2 * kh;
  const float* k1 = k0 + (size_t)16 * DH_;
  v8f c00 = {}, c01 = {}, c10 = {}, c11 = {};
#pragma unroll
  for (int kb = 0; kb < D_; kb += 4) {
    const v2f a0 = *(const v2f*)(q0 + kb);
    const v2f a1 = *(const v2f*)(q1 + kb);
    const v2f b0 = *(const v2f*)(k0 + kb);   // B[d][j] = K[j][d]
    const v2f b1 = *(const v2f*)(k1 + kb);
    c00 = wmma_f32(a0, b0, c00);
    c01 = wmma_f32(a0, b1, c01);
    c10 = wmma_f32(a1, b0, c10);
    c11 = wmma_f32(a1, b1, c11);
  }
  const size_t base = (((size_t)bh * S1_) + i0 + kh * 8) * S2_ + j0 + r;
#pragma unroll
  for (int v = 0; v < 8; ++v) {
    const size_t i00 = base + (size_t)v * S2_;
    const size_t i10 = i00 + (size_t)16 * S2_;
    abs_o[i00]      = fabsf(c00[v]);
    abs_o[i00 + 16] = fabsf(c01[v]);
    abs_o[i10]      = fabsf(c10[v]);
    abs_o[i10 + 16] = fabsf(c11[v]);
    if (WRITE_SIGNED) {
      sgn[i00]      = c00[v];
      sgn[i00 + 16] = c01[v];
      sgn[i10]      = c10[v];
      sgn[i10 + 16] = c11[v];
    }
  }
}

// in-place row softmax of (scale*s + mask*-1e9) over the last dim (S1_ = 1024)
__global__ void softmax_mask_kernel(float* __restrict__ logits,
                                    const int* __restrict__ mask,
                                    float scale) {
  const int row = blockIdx.x;            // 0 .. B_*H_*S1_-1
  const int i   = row & (S1_ - 1);
  const int b   = row >> 14;             // / (H_*S1_)
  float* rp = logits + (size_t)row * S1_;
  const int* mp = mask + ((size_t)b * S1_ + i) * S1_;   // broadcast over h
  const int t = threadIdx.x;             // 256 threads, 4 elems each
  float vals[4];
  float mx = -3.4e38f;
#pragma unroll
  for (int e = 0; e < 4; ++e) {
    const int j = t + e * 256;
    const float l = rp[j] * scale + (float)mp[j] * -1e9f;
    vals[e] = l;
    mx = fmaxf(mx, l);
  }
  __shared__ float red[256];
  red[t] = mx;
  __syncthreads();
  for (int s = 128; s > 0; s >>= 1) {
    if (t < s) red[t] = fmaxf(red[t], red[t + s]);
    __syncthreads();
  }
  mx = red[0];
  __syncthreads();
  float sum = 0.f;
#pragma unroll
  for (int e = 0; e < 4; ++e) {
    vals[e] = __expf(vals[e] - mx);
    sum += vals[e];
  }
  red[t] = sum;
  __syncthreads();
  for (int s = 128; s > 0; s >>= 1) {
    if (t < s) red[t] += red[t + s];
    __syncthreads();
  }
  const float inv = 1.0f / red[0];
#pragma unroll
  for (int e = 0; e < 4; ++e)
    rp[t + e * 256] = vals[e] * inv;
}

// concat[b,i,h*64+d] = sum_j attw[bh,i,j] * V[b,j,h,d]; 2 i-tiles x 2 d-tiles/wave
__global__ void av_kernel(const float* __restrict__ attw,
                          const float* __restrict__ Vf,
                          float* __restrict__ concat) {
  const int lane = threadIdx.x;
  const int r    = lane & 15;
  const int kh   = lane >> 4;
  const int bh   = blockIdx.z;
  const int b    = bh >> 4;
  const int h    = bh & 15;
  const int i0   = blockIdx.x * 32;
  const int d0   = threadIdx.y * 32;     // 2 waves cover 64 head cols (2 tiles each)
  const float* a0r = attw + ((size_t)bh * S1_ + i0 + r) * S1_ + 2 * kh;
  const float* a1r = a0r + (size_t)16 * S1_;
  const float* vc  = Vf + (size_t)b * S1_ * DH_ + (size_t)(2 * kh) * DH_ + h * D_ + d0 + r;
  v8f c00 = {}, c01 = {}, c10 = {}, c11 = {};
#pragma unroll 8
  for (int kb = 0; kb < S1_; kb += 4) {
    const v2f a0 = *(const v2f*)(a0r + kb);
    const v2f a1 = *(const v2f*)(a1r + kb);
    v2f b0, b1;
    b0.x = vc[kb * DH_];                 // B[j][d] = V[b,j,h,d]
    b0.y = vc[kb * DH_ + DH_];
    b1.x = vc[kb * DH_ + 16];
    b1.y = vc[kb * DH_ + DH_ + 16];
    c00 = wmma_f32(a0, b0, c00);
    c01 = wmma_f32(a0, b1, c01);
    c10 = wmma_f32(a1, b0, c10);
    c11 = wmma_f32(a1, b1, c11);
  }
  float* o0 = concat + (size_t)(b * S1_ + i0 + kh * 8) * DH_ + h * D_ + d0 + r;
  float* o1 = o0 + (size_t)16 * DH_;
#pragma unroll
  for (int v = 0; v < 8; ++v) {
    o0[(size_t)v * DH_]      = c00[v];
    o0[(size_t)v * DH_ + 16] = c01[v];
    o1[(size_t)v * DH_]      = c10[v];
    o1[(size_t)v * DH_ + 16] = c11[v];
  }
}

extern "C" void kernel_launch(void* const* d_in, const int* in_sizes, int n_in,
                              void* d_out, int out_size, void* d_ws, size_t ws_size,
                              hipStream_t stream) {
  (void)in_sizes; (void)n_in; (void)out_size; (void)ws_size;
  // Input order: x1, x2, then params in insertion order (w,b pairs), then mask.
  const float* x1      = (const float*)d_in[0];
  const float* x2      = (const float*)d_in[1];
  const float* wq1_w   = (const float*)d_in[2];
  const float* wq1_b   = (const float*)d_in[3];
  const float* wk1_w   = (const float*)d_in[4];
  const float* wk1_b   = (const float*)d_in[5];
  const float* wv1_w   = (const float*)d_in[6];
  const float* wv1_b   = (const float*)d_in[7];
  const float* wq12_w  = (const float*)d_in[8];
  const float* wq12_b  = (const float*)d_in[9];
  const float* wk12_w  = (const float*)d_in[10];
  const float* wk12_b  = (const float*)d_in[11];
  // d_in[12..17] = wq2/wk2/wv2 (unused by the reference forward)
  const float* dense1_w = (const float*)d_in[18];
  const float* dense1_b = (const float*)d_in[19];
  const float* dense2_w = (const float*)d_in[20];
  const float* dense2_b = (const float*)d_in[21];
  const int*   mask     = (const int*)d_in[22];

  float* out       = (float*)d_out;
  float* out_attn1 = out;                         // (2,1024,64)
  float* out_ff2   = out + 131072;                // (2,1024,64)
  float* attw      = out + 262144;                // (2,16,1024,1024) — scratch for signed scores
  float* attn1     = attw + 33554432;             // |scores1|
  float* attn12    = attn1 + 33554432;            // |scores12|

  float* ws  = (float*)d_ws;                      // 6 x 2M floats = 48 MB
  float* Q1  = ws;
  float* K1  = Q1  + 2097152;
  float* V1  = K1  + 2097152;
  float* Q12 = V1  + 2097152;
  float* K12 = Q12 + 2097152;                     // == k12_flat layout
  float* CC  = K12 + 2097152;                     // concat heads

  // 1) projections: (2048x64) @ (64x1024) + bias; 32x32 per wave
  const dim3 blk4(32, 4);
  const dim3 gproj(64, 8);                        // 64 m-pairs x 32 n-pairs / 4 waves
  gemm_bias_kernel<1024, 64><<<gproj, blk4, 0, stream>>>(x1, wq1_w,  wq1_b,  Q1);
  gemm_bias_kernel<1024, 64><<<gproj, blk4, 0, stream>>>(x1, wk1_w,  wk1_b,  K1);
  gemm_bias_kernel<1024, 64><<<gproj, blk4, 0, stream>>>(x1, wv1_w,  wv1_b,  V1);
  gemm_bias_kernel<1024, 64><<<gproj, blk4, 0, stream>>>(x1, wq12_w, wq12_b, Q12);
  gemm_bias_kernel<1024, 64><<<gproj, blk4, 0, stream>>>(x2, wk12_w, wk12_b, K12);

  // 2) scores: signed -> attw (scratch), abs -> attn1 / attn12
  const dim3 gsc(32, 8, 32);                      // 32 i-pairs, 32 j-pairs / 4 waves, B*H
  scores_kernel<true ><<<gsc, blk4, 0, stream>>>(Q1,  K1,  attw, attn1);
  scores_kernel<false><<<gsc, blk4, 0, stream>>>(Q12, K12, attw, attn12);

  // 3) softmax(scale*s + mask*-1e9) in place over attw
  softmax_mask_kernel<<<32768, 256, 0, stream>>>(attw, mask, 0.125f);

  // 4) attention @ V -> concat heads
  av_kernel<<<dim3(32, 1, 32), dim3(32, 2), 0, stream>>>(attw, V1, CC);

  // 5) output projections: (2048x1024) @ (1024x64) + bias
  const dim3 blk2(32, 2);
  gemm_bias_kernel<64, 1024><<<dim3(64, 1), blk2, 0, stream>>>(CC,  dense1_w, dense1_b, out_attn1);
  gemm_bias_kernel<64, 1024><<<dim3(64, 1), blk2, 0, stream>>>(K12, dense2_w, dense2_b, out_ff2);
}